// SplineConv_25563645346660
// MI455X (gfx1250) — compile-verified
//
#include <hip/hip_runtime.h>
#include <hip/hip_bf16.h>

// Problem constants (from reference)
#define NNODES 50000
#define NEDGES 1600000
#define KS0    5
#define KBINS  25          // 5*5
#define FIN    32
#define FOUT   64
#define ACCK   (KBINS*FIN) // 800
#define CHUNK  32          // K-floats staged per LDS round
#define NCHUNK 26          // 25 acc chunks + 1 x chunk (weight is contiguous (26,32,64))
#define LDS_PITCH 36       // floats; 16 distinct banks across rows, 16B-aligned pitch

typedef __attribute__((ext_vector_type(2))) float v2f;
typedef __attribute__((ext_vector_type(8))) float v8f;

// ---------------------------------------------------------------------------
// Kernel 1: zero the accumulator (N*25*32 f32) and degree (N f32) in d_ws
// ---------------------------------------------------------------------------
__global__ void zero_ws_kernel(float4* __restrict__ ws, int n4) {
    int i = blockIdx.x * blockDim.x + threadIdx.x;
    if (i < n4) ws[i] = make_float4(0.f, 0.f, 0.f, 0.f);
}

// ---------------------------------------------------------------------------
// Kernel 2: edge scatter. One wave32 per edge; lane = input-feature channel.
// Coalesced gather of x[col], coalesced global_atomic_add_f32 into acc.
// ---------------------------------------------------------------------------
__global__ void edge_scatter_kernel(const float* __restrict__ x,
                                    const float* __restrict__ pseudo,
                                    const int*   __restrict__ ei,
                                    float* __restrict__ acc,
                                    float* __restrict__ deg) {
    int gid  = blockIdx.x * blockDim.x + threadIdx.x;
    int e    = gid >> 5;
    int lane = gid & 31;
    if (e >= NEDGES) return;

    int row = ei[e];
    int col = ei[NEDGES + e];

    float p0 = pseudo[2 * e + 0];
    float p1 = pseudo[2 * e + 1];
    float v0 = p0 * (float)(KS0 - 1);      // open spline: v = p*(ks-1)
    float v1 = p1 * (float)(KS0 - 1);
    float b0 = floorf(v0), b1 = floorf(v1);
    float f0 = v0 - b0,    f1 = v1 - b1;
    float g0 = 1.0f - f0,  g1 = 1.0f - f1;
    int i0 = (int)b0, i1 = (int)b1;
    int i0a = min(i0,     KS0 - 1), i0b = min(i0 + 1, KS0 - 1);
    int i1a = min(i1,     KS0 - 1), i1b = min(i1 + 1, KS0 - 1);

    float xi = x[(size_t)col * FIN + lane];
    if (lane == 0) atomicAdd(&deg[row], 1.0f);

    float* accrow = acc + (size_t)row * ACCK + lane;
    atomicAdd(accrow + (size_t)(i0a + KS0 * i1a) * FIN, g0 * g1 * xi);
    atomicAdd(accrow + (size_t)(i0a + KS0 * i1b) * FIN, g0 * f1 * xi);
    atomicAdd(accrow + (size_t)(i0b + KS0 * i1a) * FIN, f0 * g1 * xi);
    atomicAdd(accrow + (size_t)(i0b + KS0 * i1b) * FIN, f0 * f1 * xi);
}

// ---------------------------------------------------------------------------
// Kernel 3: LDS-staged tiled GEMM with V_WMMA_F32_16X16X4_F32.
// One 16-row tile per block; 4 waves = the 4 column tiles of OUT=64.
// The 16x32 A chunk (shared by all 4 waves) is staged through LDS once per
// block instead of 4x per-lane strided loads -> ~4x less L2 read traffic on
// the 160MB accumulator (which is L2-resident: 160MB < 192MB global L2).
//
// ISA operand layouts (32-bit WMMA):
//   A 16x4 : lane -> M = lane&15, VGPR v -> K = v + 2*(lane>>4)
//   B 4x16 : lane -> N = lane&15, VGPR v -> K = v + 2*(lane>>4)
//   C 16x16: lane -> N = lane&15, VGPR j -> M = j + 8*(lane>>4)
// LDS pitch 36 floats: r*36 mod 64 distinct for r=0..15 (conflict-free b64
// reads; the two K-halves hit banks ===0 vs ===2 mod 4), 144B pitch keeps b128
// stores and b64 reads aligned.
// ---------------------------------------------------------------------------
__global__ void __launch_bounds__(128)
spline_gemm_kernel(const float* __restrict__ acc,
                   const float* __restrict__ x,
                   const float* __restrict__ weight, // (26, 32, 64) row-major
                   const float* __restrict__ bias,
                   const float* __restrict__ deg,
                   float* __restrict__ out) {
    __shared__ float lds[16 * LDS_PITCH];

    const int tid  = threadIdx.x;
    const int lane = tid & 31;
    const int wv   = tid >> 5;             // 0..3 -> output column tile
    const int cb   = wv * 16;
    const int base = blockIdx.x * 16;      // row-tile base (N % 16 == 0)
    const int nl   = lane & 15;
    const int kh   = lane >> 4;            // K half select (0/1)

    // cooperative staging role: thread t loads A[srow][skq..skq+3]
    const int srow = tid >> 3;             // 0..15
    const int skq  = (tid & 7) * 4;        // 0,4,...,28

    v8f c = {0.f, 0.f, 0.f, 0.f, 0.f, 0.f, 0.f, 0.f};

    // prefetch chunk 0 of A (acc rows) into registers
    float4 stage = *(const float4*)(acc + (size_t)(base + srow) * ACCK + skq);

    for (int ch = 0; ch < NCHUNK; ++ch) {
        __syncthreads();                   // previous compute done reading LDS
        *(float4*)(lds + srow * LDS_PITCH + skq) = stage;
        __syncthreads();                   // staged chunk visible

        // prefetch next chunk while this one is consumed
        if (ch + 1 < NCHUNK) {
            if (ch + 1 < NCHUNK - 1) {     // next is an acc chunk
                stage = *(const float4*)(acc + (size_t)(base + srow) * ACCK
                                             + (ch + 1) * CHUNK + skq);
            } else {                       // final chunk: root term A = x (N,32)
                stage = *(const float4*)(x + (size_t)(base + srow) * FIN + skq);
            }
        }

        // 8 WMMA k-steps over this 32-wide chunk
        const float* wb = weight + (size_t)ch * CHUNK * FOUT;  // chunk 25 == weight[25]
        const float* arow = lds + nl * LDS_PITCH + 2 * kh;
#pragma unroll
        for (int k = 0; k < CHUNK; k += 4) {
            v2f a = *(const v2f*)(arow + k);          // ds_load_b64, conflict-free
            v2f b;
            b[0] = wb[(size_t)(k + 2 * kh)     * FOUT + cb + nl];
            b[1] = wb[(size_t)(k + 2 * kh + 1) * FOUT + cb + nl];
            c = __builtin_amdgcn_wmma_f32_16x16x4_f32(
                    /*neg_a=*/false, a, /*neg_b=*/false, b,
                    /*c_mod=*/(short)0, c, /*reuse_a=*/false, /*reuse_b=*/false);
        }

        // after the last acc chunk (before the root term): per-row mean 1/deg
        if (ch == NCHUNK - 2) {
#pragma unroll
            for (int j = 0; j < 8; ++j) {
                float d = deg[base + j + 8 * kh];
                c[j] *= 1.0f / fmaxf(d, 1.0f);
            }
        }
    }

    // ---- bias + store ----
    const float bv = bias[cb + nl];
#pragma unroll
    for (int j = 0; j < 8; ++j) {
        out[(size_t)(base + j + 8 * kh) * FOUT + cb + nl] = c[j] + bv;
    }
}

// ---------------------------------------------------------------------------
// Launcher
// ---------------------------------------------------------------------------
extern "C" void kernel_launch(void* const* d_in, const int* in_sizes, int n_in,
                              void* d_out, int out_size, void* d_ws, size_t ws_size,
                              hipStream_t stream) {
    const float* x      = (const float*)d_in[0];   // (N, 32)
    const float* pseudo = (const float*)d_in[1];   // (E, 2)
    const int*   ei     = (const int*)  d_in[2];   // (2, E)
    const float* weight = (const float*)d_in[3];   // (26, 32, 64)
    const float* bias   = (const float*)d_in[4];   // (64,)
    float*       out    = (float*)d_out;           // (N, 64)

    float* acc = (float*)d_ws;                           // N*25*32 f32 (160 MB)
    float* deg = acc + (size_t)NNODES * KBINS * FIN;     // N f32

    // zero acc + deg
    const int zfloats = NNODES * ACCK + NNODES;          // divisible by 4
    const int z4      = zfloats / 4;
    zero_ws_kernel<<<(z4 + 255) / 256, 256, 0, stream>>>((float4*)d_ws, z4);

    // edge scatter: one wave32 per edge
    const long long threads = (long long)NEDGES * 32;
    edge_scatter_kernel<<<(int)((threads + 255) / 256), 256, 0, stream>>>(
        x, pseudo, ei, acc, deg);

    // WMMA GEMM: 3125 row tiles x (4 waves per block for the 4 col tiles)
    spline_gemm_kernel<<<NNODES / 16, 128, 0, stream>>>(
        acc, x, weight, bias, deg, out);
}